// RecurrentPreisachLayer_61022895341896
// MI455X (gfx1250) — compile-verified
//
#include <hip/hip_runtime.h>

// Problem constants (reference: T=4096, D=8192, float32)
constexpr int T  = 4096;
constexpr int D  = 8192;
constexpr int C  = 16;        // chunks over the time axis
constexpr int L  = T / C;     // 256 steps per chunk (EVEN -> chunk map slope = +1)
constexpr int S4 = D / 4;     // row stride in float4 units (2048)
constexpr int PF = 32;        // prefetch distance in rows
constexpr int DEPTH = 8;      // async-LDS pipeline depth (power of two)

typedef float v4f __attribute__((ext_vector_type(4)));

// ---- gfx1250 async global->LDS path via inline asm ------------------------
// The clang builtin exists but takes LangAS::cuda_device-qualified v4i*
// params that cannot be spelled portably from HIP; use the ISA directly.
#if defined(__gfx1250__)
#define USE_ASYNC_LDS 1
#else
#define USE_ASYNC_LDS 0
#endif

#if USE_ASYNC_LDS
__device__ __forceinline__ void async_cp16(const void* gsrc, const void* ldst) {
  // VDST = per-lane LDS byte address (low 32 bits of generic LDS pointer,
  // per the flat-addressing truncation rule); VADDR = 64-bit global address.
  asm volatile("global_load_async_to_lds_b128 %0, %1, off"
               :
               : "v"((unsigned int)(unsigned long long)ldst),
                 "v"((unsigned long long)gsrc)
               : "memory");
}
#endif

template <int N>
__device__ __forceinline__ void wait_async() {
#if defined(__gfx1250__) && __has_builtin(__builtin_amdgcn_s_wait_asynccnt)
  __builtin_amdgcn_s_wait_asynccnt(N);
#elif defined(__gfx1250__)
  asm volatile("s_wait_asynccnt %0" ::"n"(N) : "memory");
#endif
}

// ---------------------------------------------------------------------------
__device__ __forceinline__ float clip1(float z) {
  return fminf(1.0f, fmaxf(-1.0f, z));
}

// Compose one step map  M(e) = clip(a - e, -1, 1)  onto the running map
// h(e) = clip(s*e + c, lo, hi)   (slope s tracked implicitly by parity):
//   c' = a - c ; lo' = clip(a - hi) ; hi' = clip(a - lo)
__device__ __forceinline__ void compose(float a, float& c, float& lo, float& hi) {
  float nlo = clip1(a - hi);
  float nhi = clip1(a - lo);
  c  = a - c;
  lo = nlo;
  hi = nhi;
}

__device__ __forceinline__ void stepv(v4f& e, const v4f xt, const v4f xp) {
  e.x = clip1((xt.x - xp.x) - e.x);
  e.y = clip1((xt.y - xp.y) - e.y);
  e.z = clip1((xt.z - xp.z) - e.z);
  e.w = clip1((xt.w - xp.w) - e.w);
}

// ---------------------------------------------------------------------------
// Kernel 1: per (chunk j, 4-column group) compose the chunk's transfer map.
// Streams x once from HBM (leaves it resident in the 192MB L2).
// ---------------------------------------------------------------------------
__global__ __launch_bounds__(256)
void preisach_chunk_params(const float* __restrict__ x,
                           float* __restrict__ cA,
                           float* __restrict__ loA,
                           float* __restrict__ hiA) {
  const int col4 = blockIdx.x * blockDim.x + threadIdx.x;   // 0 .. S4-1
  const int j    = blockIdx.y;                              // chunk id
  const int t0   = j * L;
  const v4f* __restrict__ xv = (const v4f*)x;

  v4f xp;
  if (t0 == 0) xp = (v4f){0.f, 0.f, 0.f, 0.f};
  else         xp = xv[(size_t)(t0 - 1) * S4 + col4];

  float c0 = 0.f, c1 = 0.f, c2 = 0.f, c3 = 0.f;
  float l0 = -1e30f, l1 = -1e30f, l2 = -1e30f, l3 = -1e30f;
  float h0 =  1e30f, h1 =  1e30f, h2 =  1e30f, h3 =  1e30f;

#pragma unroll 4
  for (int t = t0; t < t0 + L; ++t) {
    v4f xt = xv[(size_t)t * S4 + col4];
    // Speculative prefetch (dropped by HW on invalid translation) -> no branch
    __builtin_prefetch(&xv[(size_t)(t + PF) * S4 + col4], 0, 1);
    compose(xt.x - xp.x, c0, l0, h0);
    compose(xt.y - xp.y, c1, l1, h1);
    compose(xt.z - xp.z, c2, l2, h2);
    compose(xt.w - xp.w, c3, l3, h3);
    xp = xt;
  }

  const size_t o = (size_t)j * S4 + col4;
  ((v4f*)cA)[o]  = (v4f){c0, c1, c2, c3};
  ((v4f*)loA)[o] = (v4f){l0, l1, l2, l3};
  ((v4f*)hiA)[o] = (v4f){h0, h1, h2, h3};
}

// ---------------------------------------------------------------------------
// Kernel 2: per column, fold the 16 chunk maps sequentially to obtain the
// entry state e_in for every chunk (chunk length even => slope +1):
//   e_out = clip(e_in + c, lo, hi)
// ---------------------------------------------------------------------------
__global__ __launch_bounds__(256)
void preisach_combine(const float* __restrict__ cA,
                      const float* __restrict__ loA,
                      const float* __restrict__ hiA,
                      float* __restrict__ eIn) {
  const int col4 = blockIdx.x * blockDim.x + threadIdx.x;   // 0 .. S4-1
  const v4f* __restrict__ c4  = (const v4f*)cA;
  const v4f* __restrict__ lo4 = (const v4f*)loA;
  const v4f* __restrict__ hi4 = (const v4f*)hiA;
  v4f* __restrict__ e4 = (v4f*)eIn;

  v4f e = (v4f){0.f, 0.f, 0.f, 0.f};
#pragma unroll
  for (int j = 0; j < C; ++j) {
    const size_t o = (size_t)j * S4 + col4;
    e4[o] = e;                      // entry state for chunk j
    v4f c  = c4[o];
    v4f lo = lo4[o];
    v4f hi = hi4[o];
    e.x = fminf(hi.x, fmaxf(lo.x, e.x + c.x));
    e.y = fminf(hi.y, fmaxf(lo.y, e.y + c.y));
    e.z = fminf(hi.z, fmaxf(lo.z, e.z + c.z));
    e.w = fminf(hi.w, fmaxf(lo.w, e.w + c.w));
  }
}

// ---------------------------------------------------------------------------
// Kernel 3: replay each chunk with its known entry state and emit outputs.
// x re-read hits the 192MB L2; output uses non-temporal stores so the 128MB
// output stream does not evict the L2-resident input.
//
// Async-LDS pipeline: each wave streams its OWN 512B row-slice into a
// depth-8 LDS ring via GLOBAL_LOAD_ASYNC_TO_LDS_B128.  Async loads complete
// in order, so  s_wait_asynccnt <= DEPTH-1  guarantees the oldest slot has
// landed.  Threads read only LDS bytes they themselves requested -> no
// barriers needed, per-wave ASYNCcnt discipline is sufficient.
// ---------------------------------------------------------------------------
__global__ __launch_bounds__(256)
void preisach_emit(const float* __restrict__ x,
                   const float* __restrict__ eIn,
                   float* __restrict__ out) {
  const int tid  = threadIdx.x;
  const int col4 = blockIdx.x * blockDim.x + tid;            // 0 .. S4-1
  const int j    = blockIdx.y;
  const int t0   = j * L;
  const v4f* __restrict__ xv = (const v4f*)x;
  v4f* __restrict__ ov = (v4f*)out;

  v4f e = ((const v4f*)eIn)[(size_t)j * S4 + col4];

  v4f xp;
  if (t0 == 0) xp = (v4f){0.f, 0.f, 0.f, 0.f};
  else         xp = xv[(size_t)(t0 - 1) * S4 + col4];

#if USE_ASYNC_LDS
  __shared__ v4f ring[DEPTH * 256];                          // 32 KB

  // Prologue: fill the pipeline (DEPTH rows in flight, zero VGPR cost).
#pragma unroll
  for (int p = 0; p < DEPTH; ++p)
    async_cp16(&xv[(size_t)(t0 + p) * S4 + col4], &ring[p * 256 + tid]);

  // Steady state: wait for oldest slot, consume, refill.
  for (int i = 0; i < L - DEPTH; ++i) {
    wait_async<DEPTH - 1>();
    v4f xt = ring[(i & (DEPTH - 1)) * 256 + tid];            // ds_load_b128
    stepv(e, xt, xp);
    __builtin_nontemporal_store(e, &ov[(size_t)(t0 + i) * S4 + col4]);
    async_cp16(&xv[(size_t)(t0 + i + DEPTH) * S4 + col4],
               &ring[(i & (DEPTH - 1)) * 256 + tid]);
    xp = xt;
  }

  // Drain: one full wait, then consume the last DEPTH slots.
  wait_async<0>();
#pragma unroll
  for (int k = 0; k < DEPTH; ++k) {
    const int i = L - DEPTH + k;
    v4f xt = ring[(i & (DEPTH - 1)) * 256 + tid];
    stepv(e, xt, xp);
    __builtin_nontemporal_store(e, &ov[(size_t)(t0 + i) * S4 + col4]);
    xp = xt;
  }
#else
  // Fallback: direct-load pipeline with speculative prefetch.
#pragma unroll 4
  for (int t = t0; t < t0 + L; ++t) {
    v4f xt = xv[(size_t)t * S4 + col4];
    __builtin_prefetch(&xv[(size_t)(t + PF) * S4 + col4], 0, 1);
    stepv(e, xt, xp);
    __builtin_nontemporal_store(e, &ov[(size_t)t * S4 + col4]);
    xp = xt;
  }
#endif
}

// ---------------------------------------------------------------------------
extern "C" void kernel_launch(void* const* d_in, const int* in_sizes, int n_in,
                              void* d_out, int out_size, void* d_ws, size_t ws_size,
                              hipStream_t stream) {
  (void)in_sizes; (void)n_in; (void)out_size; (void)ws_size;

  const float* x = (const float*)d_in[0];
  float* out = (float*)d_out;

  // Workspace layout (floats): c | lo | hi | e_in  -> 4 * C * D * 4B = 2 MB
  float* ws  = (float*)d_ws;
  float* cA  = ws;
  float* loA = ws + (size_t)1 * C * D;
  float* hiA = ws + (size_t)2 * C * D;
  float* eIn = ws + (size_t)3 * C * D;

  dim3 blk(256);
  dim3 gridJC(S4 / 256, C);   // (8, 16): 32768 threads, 1024 wave32s
  dim3 gridCol(S4 / 256);     // (8):     2048 threads

  preisach_chunk_params<<<gridJC, blk, 0, stream>>>(x, cA, loA, hiA);
  preisach_combine<<<gridCol, blk, 0, stream>>>(cA, loA, hiA, eIn);
  preisach_emit<<<gridJC, blk, 0, stream>>>(x, eIn, out);
}